// RealNDimFourier_36498632082014
// MI455X (gfx1250) — compile-verified
//
#include <hip/hip_runtime.h>
#include <math.h>

typedef __attribute__((ext_vector_type(2))) float v2f;
typedef __attribute__((ext_vector_type(4))) float v4f;
typedef __attribute__((ext_vector_type(8))) float v8f;

#define NPLANES 96
#define DIM 512
#define PLANE_SHIFT 18   // 512*512 = 2^18

__device__ __forceinline__ void plane_params(int i,
    const float* __restrict__ phases, const float* __restrict__ strengths,
    const int* __restrict__ idx_h, const int* __restrict__ idx_w,
    int& h0, int& w0, float& p, float& K, int& plane)
{
    h0 = idx_h[i];
    w0 = idx_w[i];
    p  = phases[i];
    const int bi = i & 31;   // i % 32
    const int ci = i % 3;
    plane = bi * 3 + ci;
    const float st = strengths[plane];
    // Degenerate self-conjugate bins (2*h == 0 mod 512 etc.)
    const float s1f = (((h0 == 0) | (h0 == 256)) & ((w0 == 0) | (w0 == 256))) ? 1.f : 0.f;
    const float s2f = (((h0 == 255) | (h0 == 511)) & ((w0 == 255) | (w0 == 511))) ? 1.f : 0.f;
    const float M = (1.f + p * p) + 0.5f * (1.f - p * p) * (s1f + s2f);
    // K = scale*strength / (HW * norm) = strength / (32*sqrt(M))
    K = st / (32.f * sqrtf(M));
}

// Precompute per-plane rank-4 factors:
//   U[i][y][0..3]  = K * { cos(a y) - p sin(a y), -(sin(a y) + p cos(a y)),  (same for a') }
//   Vt[i][x][0..3] = { cos(b x), sin(b x), cos(b' x), sin(b' x) }
__global__ void fourier_precompute(const float* __restrict__ phases,
                                   const float* __restrict__ strengths,
                                   const int* __restrict__ idx_h,
                                   const int* __restrict__ idx_w,
                                   float* __restrict__ U,
                                   float* __restrict__ Vt)
{
    const int i = blockIdx.x;   // plane 0..95
    const int t = threadIdx.x;  // coordinate 0..511
    int h0, w0, plane; float p, K;
    plane_params(i, phases, strengths, idx_h, idx_w, h0, w0, p, K, plane);

    const float C0 = 6.28318530717958647692f / 512.f;
    float s, c;

    // y-direction factors (K folded in)
    __sincosf((float)((h0 * t) & 511) * C0, &s, &c);
    const float u1 =  K * (c - p * s);
    const float u2 = -K * (s + p * c);
    __sincosf((float)(((511 - h0) * t) & 511) * C0, &s, &c);
    const float u3 =  K * (c - p * s);
    const float u4 = -K * (s + p * c);
    v4f uv = {u1, u2, u3, u4};
    *(v4f*)(U + ((size_t)(i * DIM + t) << 2)) = uv;

    // x-direction factors
    float s1, c1, s2, c2;
    __sincosf((float)((w0 * t) & 511) * C0, &s1, &c1);
    __sincosf((float)(((511 - w0) * t) & 511) * C0, &s2, &c2);
    v4f vv = {c1, s1, c2, s2};
    *(v4f*)(Vt + ((size_t)(i * DIM + t) << 2)) = vv;
}

// One wave per 16x16 output tile: D = A(16x4) * B(4x16) + C, C preloaded with x.
// A-frag  : lane<16 -> U[ty+lane][0..1], lane>=16 -> U[ty+lane-16][2..3]
// B-frag  : lane<16 -> Vt[tx+lane][0..1], lane>=16 -> Vt[tx+lane-16][2..3]
// C/D     : VGPR j <-> row (half*8 + j), col (lane&15)
__global__ void __launch_bounds__(256)
fourier_apply_wmma(const float* __restrict__ x,
                   const float* __restrict__ U,
                   const float* __restrict__ Vt,
                   float* __restrict__ out)
{
    const int wave = threadIdx.x >> 5;
    const int ln   = threadIdx.x & 31;
    const int tile = blockIdx.x * 8 + wave;   // 98304 tiles total
    const int i    = tile >> 10;              // plane index 0..95
    const int rem  = tile & 1023;
    const int ty   = (rem >> 5) << 4;
    const int tx   = (rem & 31) << 4;
    const int half = ln >> 4;
    const int col  = ln & 15;

    const int bi = i & 31;
    const int ci = i % 3;
    const size_t pbase = (size_t)(bi * 3 + ci) << PLANE_SHIFT;
    const float* __restrict__ xp = x + pbase;
    float* __restrict__ op = out + pbase;

    v2f a = *(const v2f*)(U  + ((size_t)(i * DIM + ty + col) << 2) + 2 * half);
    v2f b = *(const v2f*)(Vt + ((size_t)(i * DIM + tx + col) << 2) + 2 * half);

    const int ebase = (ty + half * 8) * DIM + tx + col;
    v8f cacc;
#pragma unroll
    for (int j = 0; j < 8; ++j) cacc[j] = xp[ebase + j * DIM];

    __builtin_prefetch(xp + ebase + 8 * DIM, 0, 1);

    v8f d = __builtin_amdgcn_wmma_f32_16x16x4_f32(
        /*neg_a=*/false, a, /*neg_b=*/false, b,
        /*c_mod=*/(short)0, cacc, /*reuse_a=*/false, /*reuse_b=*/false);

#pragma unroll
    for (int j = 0; j < 8; ++j)
        op[ebase + j * DIM] = fminf(fmaxf(d[j], 0.f), 1.f);
}

// Safety fallback (workspace too small): direct elementwise evaluation.
__global__ void fourier_fallback(const float* __restrict__ x,
                                 const float* __restrict__ phases,
                                 const float* __restrict__ strengths,
                                 const int* __restrict__ idx_h,
                                 const int* __restrict__ idx_w,
                                 float* __restrict__ out)
{
    const long long e = (long long)blockIdx.x * blockDim.x + threadIdx.x;
    const int i   = (int)(e >> PLANE_SHIFT);
    const int rem = (int)(e & ((1 << PLANE_SHIFT) - 1));
    const int y   = rem >> 9;
    const int xx  = rem & 511;
    int h0, w0, plane; float p, K;
    plane_params(i, phases, strengths, idx_h, idx_w, h0, w0, p, K, plane);

    const float C0 = 6.28318530717958647692f / 512.f;
    float s, c, wsum = 0.f;
    __sincosf((float)((h0 * y + w0 * xx) & 511) * C0, &s, &c);
    wsum += c - p * s;
    __sincosf((float)(((511 - h0) * y + (511 - w0) * xx) & 511) * C0, &s, &c);
    wsum += c - p * s;

    const size_t o = ((size_t)plane << PLANE_SHIFT) + rem;
    const float v = x[o] + K * wsum;
    out[o] = fminf(fmaxf(v, 0.f), 1.f);
}

extern "C" void kernel_launch(void* const* d_in, const int* in_sizes, int n_in,
                              void* d_out, int out_size, void* d_ws, size_t ws_size,
                              hipStream_t stream) {
    (void)in_sizes; (void)n_in; (void)out_size;
    const float* x         = (const float*)d_in[0];
    const float* phases    = (const float*)d_in[1];
    const float* strengths = (const float*)d_in[2];
    const int*   idx_h     = (const int*)d_in[3];
    const int*   idx_w     = (const int*)d_in[4];
    float*       out       = (float*)d_out;

    const size_t need = (size_t)NPLANES * DIM * 4 * 2 * sizeof(float); // ~1.5 MB
    if (d_ws != nullptr && ws_size >= need) {
        float* U  = (float*)d_ws;
        float* Vt = U + (size_t)NPLANES * DIM * 4;
        fourier_precompute<<<NPLANES, DIM, 0, stream>>>(phases, strengths, idx_h, idx_w, U, Vt);
        // 96 planes * 32 * 32 tiles = 98304 tiles, 8 waves (tiles) per 256-thread block
        fourier_apply_wmma<<<98304 / 8, 256, 0, stream>>>(x, U, Vt, out);
    } else {
        fourier_fallback<<<98304, 256, 0, stream>>>(x, phases, strengths, idx_h, idx_w, out);
    }
}